// KMeansLayer_15341623181312
// MI455X (gfx1250) — compile-verified
//
#include <hip/hip_runtime.h>

// ---------------------------------------------------------------------------
// K-means (N=65536, D=256, K=512, 10 iters) for gfx1250 / MI455X.
// Distance matmul on V_WMMA_F32_16X16X4_F32 (fp32 matrix pipe, wave32).
// Dataset (64MB) + centers (512KB) fit in the 192MB L2 -> compute bound.
// Each wave drives TWO independent WMMA accumulator chains sharing the A
// fragment, to keep the matrix pipe full (no single-chain RAW stalls).
// ---------------------------------------------------------------------------

typedef float v2f __attribute__((ext_vector_type(2)));
typedef float v8f __attribute__((ext_vector_type(8)));

constexpr int Nn    = 65536;
constexpr int Dd    = 256;
constexpr int Kk    = 512;
constexpr int NITER = 10;

// ---------------------------------------------------------------------------
// Per-sample squared norms: one wave (32 lanes) per row, 8 rows per block.
// ---------------------------------------------------------------------------
__global__ __launch_bounds__(256) void x2_kernel(const float* __restrict__ data,
                                                 float* __restrict__ x2) {
  const int t = threadIdx.x;
  const int wave = t >> 5, lane = t & 31;
  const int row = blockIdx.x * 8 + wave;
  const float* p = data + row * Dd;
  float v = 0.f;
#pragma unroll
  for (int q = 0; q < Dd / 32; ++q) {
    float x = p[lane + 32 * q];
    v += x * x;
  }
#pragma unroll
  for (int off = 16; off > 0; off >>= 1) v += __shfl_xor(v, off, 32);
  if (lane == 0) x2[row] = v;
}

// ---------------------------------------------------------------------------
// Gather initial centers and compute center norms. One block per center.
// ---------------------------------------------------------------------------
__global__ __launch_bounds__(256) void init_centers_kernel(
    const float* __restrict__ data, const int* __restrict__ init_idx,
    float* __restrict__ c, float* __restrict__ c2) {
  const int k = blockIdx.x, d = threadIdx.x;  // d in [0,256) == D
  const float v = data[init_idx[k] * Dd + d];
  c[k * Dd + d] = v;
  __shared__ float red[256];
  red[d] = v * v;
  __syncthreads();
  for (int s = 128; s > 0; s >>= 1) {
    if (d < s) red[d] += red[d + s];
    __syncthreads();
  }
  if (d == 0) c2[k] = red[0];
}

// ---------------------------------------------------------------------------
// Zero sums[K*D] and counts[K].
// ---------------------------------------------------------------------------
__global__ __launch_bounds__(256) void zero_kernel(float* __restrict__ sums,
                                                   float* __restrict__ counts) {
  const int gid = blockIdx.x * 256 + threadIdx.x;
  if (gid < Kk * Dd) sums[gid] = 0.f;
  else if (gid < Kk * Dd + Kk) counts[gid - Kk * Dd] = 0.f;
}

// ---------------------------------------------------------------------------
// Assignment + scatter accumulation.
// Block = 16 samples x all 512 centers. 8 waves; each wave processes 2
// pairs of 16-center tiles; within a pair, two WMMA chains run concurrently
// sharing the A (sample) fragment loaded from LDS.
// Scores: ||c||^2 - 2 x.c  (||x||^2 constant per row -> irrelevant to argmin)
// ---------------------------------------------------------------------------
__global__ __launch_bounds__(256) void assign_kernel(
    const float* __restrict__ data, const float* __restrict__ c,
    const float* __restrict__ c2, int* __restrict__ a,
    float* __restrict__ sums, float* __restrict__ counts) {
  __shared__ float sA[16 * Dd];       // 16 KB: sample tile
  __shared__ float sScore[16 * Kk];   // 32 KB: 16 x 512 scores
  __shared__ float sBestVal[16 * 16];
  __shared__ int   sBestIdx[16 * 16];

  const int m0 = blockIdx.x * 16;
  const int t = threadIdx.x;

  // Stage the 16 x 256 sample tile into LDS (coalesced, 16 floats/thread).
  for (int i = t; i < 16 * Dd; i += 256)
    sA[i] = data[(m0 + (i >> 8)) * Dd + (i & (Dd - 1))];
  __syncthreads();

  const int wave = t >> 5;
  const int lane = t & 31;
  const int lrow = lane & 15;   // M (A) / N column (B)
  const int hi   = lane >> 4;   // lane-half -> K offset pairs

  for (int j = 0; j < 2; ++j) {
    const int n0a = wave * 16 + j * 128;   // first center tile
    const int n0b = n0a + 256;             // second center tile
    const float* crowA = c + (n0a + lrow) * Dd;
    const float* crowB = c + (n0b + lrow) * Dd;
    __builtin_prefetch(crowA, 0, 1);       // global_prefetch_b8
    __builtin_prefetch(crowB, 0, 1);
    v8f accA = {};
    v8f accB = {};
#pragma unroll 4
    for (int k = 0; k < Dd; k += 4) {
      v2f av, bva, bvb;
      av.x = sA[lrow * Dd + k + hi * 2 + 0];   // shared A: 16x4 samples
      av.y = sA[lrow * Dd + k + hi * 2 + 1];
      bva.x = crowA[k + hi * 2 + 0];           // B tile a (c^T)
      bva.y = crowA[k + hi * 2 + 1];
      bvb.x = crowB[k + hi * 2 + 0];           // B tile b (c^T)
      bvb.y = crowB[k + hi * 2 + 1];
      accA = __builtin_amdgcn_wmma_f32_16x16x4_f32(
          false, av, false, bva, (short)0, accA, false, false);
      accB = __builtin_amdgcn_wmma_f32_16x16x4_f32(
          false, av, false, bvb, (short)0, accB, false, false);
    }
    // C layout: VGPR r, lanes 0-15 -> M=r ; lanes 16-31 -> M=r+8 ; N=lane&15
    const int nca = n0a + lrow;
    const int ncb = n0b + lrow;
    const float cna = c2[nca];
    const float cnb = c2[ncb];
    const int mbase = hi * 8;
#pragma unroll
    for (int r = 0; r < 8; ++r) {
      sScore[(mbase + r) * Kk + nca] = cna - 2.0f * accA[r];
      sScore[(mbase + r) * Kk + ncb] = cnb - 2.0f * accB[r];
    }
  }
  __syncthreads();

  // Block argmin: 16 threads per sample, 32 candidates each, then combine.
  const int s = t >> 4, ch = t & 15;
  float best = 3.4e38f;
  int bidx = 0;
  for (int q = 0; q < 32; ++q) {
    const int k = ch * 32 + q;
    const float v = sScore[s * Kk + k];
    if (v < best) { best = v; bidx = k; }
  }
  sBestVal[s * 16 + ch] = best;
  sBestIdx[s * 16 + ch] = bidx;
  __syncthreads();
  if (ch == 0) {  // first-occurrence tie-break: chunks scanned in k order
    float bb = sBestVal[s * 16];
    int bi = sBestIdx[s * 16];
    for (int q = 1; q < 16; ++q) {
      if (sBestVal[s * 16 + q] < bb) { bb = sBestVal[s * 16 + q]; bi = sBestIdx[s * 16 + q]; }
    }
    a[m0 + s] = bi;
    atomicAdd(&counts[bi], 1.0f);
    sBestIdx[s * 16] = bi;  // broadcast winner
  }
  __syncthreads();

  // Scatter-add: each thread owns one column d=t, loops 16 samples.
  for (int s2 = 0; s2 < 16; ++s2) {
    const int bi = sBestIdx[s2 * 16];
    atomicAdd(&sums[bi * Dd + t], sA[s2 * Dd + t]);
  }
}

// ---------------------------------------------------------------------------
// Center update: mean or dead-cluster reinit; also new center norms.
// ---------------------------------------------------------------------------
__global__ __launch_bounds__(256) void update_kernel(
    const float* __restrict__ sums, const float* __restrict__ counts,
    const float* __restrict__ data, const int* __restrict__ reinit, int iter,
    float* __restrict__ c, float* __restrict__ c2) {
  const int k = blockIdx.x, d = threadIdx.x;
  const float cnt = counts[k];
  float v;
  if (cnt == 0.0f)
    v = data[reinit[iter * Kk + k] * Dd + d];
  else
    v = sums[k * Dd + d] / fmaxf(cnt, 1.0f);
  c[k * Dd + d] = v;
  __shared__ float red[256];
  red[d] = v * v;
  __syncthreads();
  for (int s = 128; s > 0; s >>= 1) {
    if (d < s) red[d] += red[d + s];
    __syncthreads();
  }
  if (d == 0) c2[k] = red[0];
}

// ---------------------------------------------------------------------------
// Final pass: nearest data point per center (argmin over N).
// Block = 16 centers; 8 waves sweep N in pairs of 16-sample tiles with two
// concurrent WMMA chains sharing the center (A) fragment.
// score = ||x||^2 - 2 x.c  (||c||^2 constant per center row).
// ---------------------------------------------------------------------------
__global__ __launch_bounds__(256) void final_kernel(
    const float* __restrict__ data, const float* __restrict__ c,
    const float* __restrict__ x2, int* __restrict__ indexOut) {
  __shared__ float sC[16 * Dd];       // 16 KB: center tile
  __shared__ float rV[16 * 128];      // 8 KB: per-(center,slot) best value
  __shared__ int   rN[16 * 128];      // 8 KB: per-(center,slot) best index
  __shared__ float pV[16 * 16];
  __shared__ int   pN[16 * 16];

  const int k0 = blockIdx.x * 16;
  const int t = threadIdx.x;
  for (int i = t; i < 16 * Dd; i += 256)
    sC[i] = c[(k0 + (i >> 8)) * Dd + (i & (Dd - 1))];
  __syncthreads();

  const int wave = t >> 5;
  const int lane = t & 31;
  const int lrow = lane & 15;
  const int hi   = lane >> 4;

  float bestv[8];
  int   bestn[8];
#pragma unroll
  for (int r = 0; r < 8; ++r) { bestv[r] = 3.4e38f; bestn[r] = 0; }

  for (int jj = 0; jj < Nn / 256; ++jj) {
    const int n0a = jj * 256 + wave * 16;   // first sample tile
    const int n0b = n0a + 128;              // second sample tile
    const int nsa = n0a + lrow;
    const int nsb = n0b + lrow;
    const float* drowA = data + nsa * Dd;
    const float* drowB = data + nsb * Dd;
    v8f accA = {};
    v8f accB = {};
#pragma unroll 4
    for (int k = 0; k < Dd; k += 4) {
      v2f av, bva, bvb;
      av.x = sC[lrow * Dd + k + hi * 2 + 0];  // shared A: 16 centers x 4
      av.y = sC[lrow * Dd + k + hi * 2 + 1];
      bva.x = drowA[k + hi * 2 + 0];          // B tile a (data^T)
      bva.y = drowA[k + hi * 2 + 1];
      bvb.x = drowB[k + hi * 2 + 0];          // B tile b (data^T)
      bvb.y = drowB[k + hi * 2 + 1];
      accA = __builtin_amdgcn_wmma_f32_16x16x4_f32(
          false, av, false, bva, (short)0, accA, false, false);
      accB = __builtin_amdgcn_wmma_f32_16x16x4_f32(
          false, av, false, bvb, (short)0, accB, false, false);
    }
    const float xna = x2[nsa];
    const float xnb = x2[nsb];
#pragma unroll
    for (int r = 0; r < 8; ++r) {
      const float va = xna - 2.0f * accA[r];
      if (va < bestv[r] || (va == bestv[r] && nsa < bestn[r])) {
        bestv[r] = va;
        bestn[r] = nsa;
      }
      const float vb = xnb - 2.0f * accB[r];
      if (vb < bestv[r] || (vb == bestv[r] && nsb < bestn[r])) {
        bestv[r] = vb;
        bestn[r] = nsb;
      }
    }
  }

  // Dump per-lane running bests: center m = hi*8 + r, slot = wave*16 + lrow.
  const int slot = wave * 16 + lrow;
#pragma unroll
  for (int r = 0; r < 8; ++r) {
    const int m = hi * 8 + r;
    rV[m * 128 + slot] = bestv[r];
    rN[m * 128 + slot] = bestn[r];
  }
  __syncthreads();

  // Reduce 128 slots -> 16 -> 1 per center (tie-break: smaller sample index).
  const int m = t >> 4, part = t & 15;
  float bb = 3.4e38f;
  int bi = 0;
  for (int q = 0; q < 8; ++q) {
    const float v = rV[m * 128 + part * 8 + q];
    const int n = rN[m * 128 + part * 8 + q];
    if (v < bb || (v == bb && n < bi)) { bb = v; bi = n; }
  }
  pV[m * 16 + part] = bb;
  pN[m * 16 + part] = bi;
  __syncthreads();
  if (part == 0) {
    float fb = pV[m * 16];
    int fi = pN[m * 16];
    for (int q = 1; q < 16; ++q) {
      const float v = pV[m * 16 + q];
      const int n = pN[m * 16 + q];
      if (v < fb || (v == fb && n < fi)) { fb = v; fi = n; }
    }
    indexOut[k0 + m] = fi;
  }
}

// ---------------------------------------------------------------------------
// Pack outputs: [avg_center K*D][a N][center K*D][index K], ints as floats.
// ---------------------------------------------------------------------------
__global__ __launch_bounds__(256) void writeout_kernel(
    const float* __restrict__ c, const int* __restrict__ a,
    const int* __restrict__ idx, const float* __restrict__ data,
    float* __restrict__ out) {
  const int gid = blockIdx.x * 256 + threadIdx.x;
  const int KD = Kk * Dd;
  const int total = KD + Nn + KD + Kk;
  if (gid >= total) return;
  if (gid < KD) {
    out[gid] = c[gid];
  } else if (gid < KD + Nn) {
    out[gid] = (float)a[gid - KD];
  } else if (gid < 2 * KD + Nn) {
    const int j = gid - KD - Nn;
    out[gid] = data[idx[j >> 8] * Dd + (j & (Dd - 1))];
  } else {
    out[gid] = (float)idx[gid - 2 * KD - Nn];
  }
}

// ---------------------------------------------------------------------------
extern "C" void kernel_launch(void* const* d_in, const int* in_sizes, int n_in,
                              void* d_out, int out_size, void* d_ws, size_t ws_size,
                              hipStream_t stream) {
  const float* data     = (const float*)d_in[0];
  const int*   init_idx = (const int*)d_in[1];
  const int*   reinit   = (const int*)d_in[2];
  float* out = (float*)d_out;

  // Workspace carve-up (~1.6 MB).
  float* ws     = (float*)d_ws;
  float* c      = ws;                 // K*D
  float* sums   = c + Kk * Dd;        // K*D
  float* counts = sums + Kk * Dd;     // K
  float* c2     = counts + Kk;        // K
  float* x2     = c2 + Kk;            // N
  int*   a      = (int*)(x2 + Nn);    // N
  int*   idx    = a + Nn;             // K

  x2_kernel<<<Nn / 8, 256, 0, stream>>>(data, x2);
  init_centers_kernel<<<Kk, 256, 0, stream>>>(data, init_idx, c, c2);

  for (int i = 0; i < NITER; ++i) {
    zero_kernel<<<(Kk * Dd + Kk + 255) / 256, 256, 0, stream>>>(sums, counts);
    assign_kernel<<<Nn / 16, 256, 0, stream>>>(data, c, c2, a, sums, counts);
    update_kernel<<<Kk, 256, 0, stream>>>(sums, counts, data, reinit, i, c, c2);
  }

  final_kernel<<<Kk / 16, 256, 0, stream>>>(data, c, x2, idx);

  const int total = 2 * Kk * Dd + Nn + Kk;
  writeout_kernel<<<(total + 255) / 256, 256, 0, stream>>>(c, a, idx, data, out);
}